// SplineConv_41953240547637
// MI455X (gfx1250) — compile-verified
//
#include <hip/hip_runtime.h>
#include <hip/hip_bf16.h>

// ---------------------------------------------------------------------------
// SplineConv for MI455X (gfx1250, wave32, WMMA).
//   N=20000 nodes, E=500000 edges, IN=OUT=64, KS=5, DIM=2 -> K=25 (+1 root)
// Phase 1: zero workspace (agg[N][25][64] f32 + deg[N] f32) -- 128MB, L2-resident
// Phase 2: edge scatter with global_atomic_add_f32 (L2-resident target)
// Phase 3: 32-node x 64-col tile GEMM via v_wmma_f32_16x16x4_f32.
//          Each wave owns a 16-col slice and two stacked 16x16 D tiles that
//          share the B operand (halves B traffic per WMMA). A staged in LDS
//          with 68-float pitch (conflict-free ds_load_2addr_b64 pattern).
// ---------------------------------------------------------------------------

typedef float v2f __attribute__((ext_vector_type(2)));
typedef float v8f __attribute__((ext_vector_type(8)));

#define IN_F   64
#define OUT_F  64
#define KKER   25          // spline kernels (weight[25] is the root weight)
#define MROWS  32          // node rows per block tile
#define LDS_PITCH 68       // 64 + 4 pad floats: 4-bank rotation per row

// ---------------- zero-init workspace ----------------
__global__ void zero_f32_kernel(float* __restrict__ p, size_t n) {
    size_t i = (size_t)blockIdx.x * blockDim.x + threadIdx.x;
    if (i < n) p[i] = 0.0f;
}

// ---------------- edge scatter ----------------
// 64 threads per edge: thread f handles feature f. Each computes the 4
// bilinear spline basis values redundantly (cheap VALU) and issues 4
// global_atomic_add_f32 into agg[row][kidx][f]. Lane f==0 bumps deg[row].
__global__ __launch_bounds__(256)
void spline_scatter_kernel(const float* __restrict__ x,
                           const int*   __restrict__ ei,
                           const float* __restrict__ pseudo,
                           float* __restrict__ agg,
                           float* __restrict__ deg,
                           int E) {
    int t = blockIdx.x * 256 + threadIdx.x;
    int e = t >> 6;          // edge id
    int f = t & 63;          // feature id
    if (e >= E) return;

    int row = ei[e];         // target
    int col = ei[E + e];     // source
    float xv = x[(size_t)col * IN_F + f];

    float u0 = pseudo[2 * e]     * 4.0f;   // (KS-1) = 4
    float u1 = pseudo[2 * e + 1] * 4.0f;
    int i0 = (int)floorf(u0); i0 = i0 < 0 ? 0 : (i0 > 3 ? 3 : i0);
    int i1 = (int)floorf(u1); i1 = i1 < 0 ? 0 : (i1 > 3 ? 3 : i1);
    float f0 = u0 - (float)i0;
    float f1 = u1 - (float)i1;

#pragma unroll
    for (int s = 0; s < 4; ++s) {
        float b0 = (s & 1) ? f0 : (1.0f - f0);
        float b1 = (s & 2) ? f1 : (1.0f - f1);
        int kidx = (i0 + (s & 1)) * 5 + (i1 + ((s >> 1) & 1));  // row-major flat
        atomicAdd(&agg[((size_t)row * KKER + kidx) * IN_F + f], b0 * b1 * xv);
    }
    if (f == 0) atomicAdd(&deg[row], 1.0f);
}

// ---------------- one 32x64(K) chunk: 32 WMMAs, B shared across M-subtiles ----
// A layout (ISA 7.12.2, 32-bit A 16x4): lane holds row M=lane&15; VGPR0/1 hold
// K = klo, klo+1 with klo = (lane>>4)*2. B (4x16): lane holds col N=lane&15,
// VGPR0/1 hold rows K = klo, klo+1.
__device__ __forceinline__
void wmma_chunk2(const float* __restrict__ wk,   // 64x64 weight slab [i][o]
                 const float* __restrict__ As,   // LDS A tile, 32 rows, pitch 68
                 int lane, int ocol, v8f& c_lo, v8f& c_hi) {
    const int m16 = lane & 15;
    const int klo = (lane >> 4) * 2;
#pragma unroll
    for (int t = 0; t < 16; ++t) {
        const int i = t * 4 + klo;
        v2f b;
        b.x = wk[(size_t)i * OUT_F + ocol];
        b.y = wk[(size_t)(i + 1) * OUT_F + ocol];
        v2f a_lo, a_hi;
        a_lo.x = As[m16 * LDS_PITCH + i];              // ds_load_2addr_b64
        a_lo.y = As[m16 * LDS_PITCH + i + 1];
        a_hi.x = As[(m16 + 16) * LDS_PITCH + i];
        a_hi.y = As[(m16 + 16) * LDS_PITCH + i + 1];
        c_lo = __builtin_amdgcn_wmma_f32_16x16x4_f32(
                   false, a_lo, false, b, (short)0, c_lo, false, false);
        c_hi = __builtin_amdgcn_wmma_f32_16x16x4_f32(
                   false, a_hi, false, b, (short)0, c_hi, false, false);
    }
}

// ---------------- GEMM: out[n,:] = (sum_k agg[n,k,:] @ W[k]) / deg + x @ W[25] + bias
// One block per 32-node tile, 128 threads = 4 waves, wave w owns cols [16w,16w+16).
__global__ __launch_bounds__(128)
void spline_gemm_kernel(const float* __restrict__ agg,
                        const float* __restrict__ x,
                        const float* __restrict__ weight,   // [26][64][64]
                        const float* __restrict__ deg,
                        const float* __restrict__ bias,
                        float* __restrict__ out) {
    __shared__ float As[MROWS * LDS_PITCH];

    const int tid  = threadIdx.x;
    const int wave = tid >> 5;
    const int lane = tid & 31;
    const int n0   = blockIdx.x * MROWS;
    const int ocol = wave * 16 + (lane & 15);

    // staging role: 128 threads x 16 floats = 32 rows x 64 cols
    const int srow = tid >> 2;            // 0..31
    const int scol = (tid & 3) * 16;      // 0,16,32,48

    v8f c_agg_lo = {}, c_agg_hi = {};
    v8f c_root_lo = {}, c_root_hi = {};

    for (int kk = 0; kk < KKER; ++kk) {
        // prefetch next chunk's weight slab into L2 (global_prefetch_b8)
        __builtin_prefetch(weight + (size_t)(kk + 1) * IN_F * OUT_F
                                  + (size_t)lane * 16, 0, 1);

        const float* src = agg + ((size_t)(n0 + srow) * KKER + kk) * IN_F + scol;
        float4 a0 = *(const float4*)(src);
        float4 a1 = *(const float4*)(src + 4);
        float4 a2 = *(const float4*)(src + 8);
        float4 a3 = *(const float4*)(src + 12);
        __syncthreads();                       // prior iteration's LDS reads done
        float* dst = &As[srow * LDS_PITCH + scol];
        *(float4*)(dst)      = a0;
        *(float4*)(dst + 4)  = a1;
        *(float4*)(dst + 8)  = a2;
        *(float4*)(dst + 12) = a3;
        __syncthreads();
        wmma_chunk2(weight + (size_t)kk * IN_F * OUT_F, As, lane, ocol,
                    c_agg_lo, c_agg_hi);
    }

    // root chunk: A = x[n0:n0+32, :], W = weight[25]
    {
        const float* src = x + (size_t)(n0 + srow) * IN_F + scol;
        float4 a0 = *(const float4*)(src);
        float4 a1 = *(const float4*)(src + 4);
        float4 a2 = *(const float4*)(src + 8);
        float4 a3 = *(const float4*)(src + 12);
        __syncthreads();
        float* dst = &As[srow * LDS_PITCH + scol];
        *(float4*)(dst)      = a0;
        *(float4*)(dst + 4)  = a1;
        *(float4*)(dst + 8)  = a2;
        *(float4*)(dst + 12) = a3;
        __syncthreads();
        wmma_chunk2(weight + (size_t)KKER * IN_F * OUT_F, As, lane, ocol,
                    c_root_lo, c_root_hi);
    }

    // epilogue: C/D layout -- VGPR v: M = v (lanes 0-15) / v+8 (lanes 16-31)
    const int mbase = (lane >> 4) * 8;
    const float bcol = bias[ocol];
#pragma unroll
    for (int v = 0; v < 8; ++v) {
        {   // rows 0..15 subtile
            int m = mbase + v;
            float inv = 1.0f / fmaxf(deg[n0 + m], 1.0f);
            out[(size_t)(n0 + m) * OUT_F + ocol] =
                c_agg_lo[v] * inv + c_root_lo[v] + bcol;
        }
        {   // rows 16..31 subtile
            int m = mbase + v + 16;
            float inv = 1.0f / fmaxf(deg[n0 + m], 1.0f);
            out[(size_t)(n0 + m) * OUT_F + ocol] =
                c_agg_hi[v] * inv + c_root_hi[v] + bcol;
        }
    }
}

// ---------------------------------------------------------------------------
extern "C" void kernel_launch(void* const* d_in, const int* in_sizes, int n_in,
                              void* d_out, int out_size, void* d_ws, size_t ws_size,
                              hipStream_t stream) {
    const float* x      = (const float*)d_in[0];   // [N,64]
    const int*   ei     = (const int*)  d_in[1];   // [2,E]
    const float* pseudo = (const float*)d_in[2];   // [E,2]
    const float* weight = (const float*)d_in[3];   // [26,64,64]
    const float* bias   = (const float*)d_in[4];   // [64]
    float*       out    = (float*)d_out;           // [N,64]

    const int N = in_sizes[0] / IN_F;              // 20000 (divisible by 32)
    const int E = in_sizes[1] / 2;                 // 500000

    float* agg = (float*)d_ws;                       // N*25*64 floats (128 MB)
    float* deg = agg + (size_t)N * KKER * IN_F;      // N floats
    size_t nzero = (size_t)N * KKER * IN_F + (size_t)N;

    zero_f32_kernel<<<(unsigned)((nzero + 255) / 256), 256, 0, stream>>>(agg, nzero);

    size_t sthreads = (size_t)E * 64;
    spline_scatter_kernel<<<(unsigned)((sthreads + 255) / 256), 256, 0, stream>>>(
        x, ei, pseudo, agg, deg, E);

    spline_gemm_kernel<<<(unsigned)(N / MROWS), 128, 0, stream>>>(
        agg, x, weight, deg, bias, out);
}